// ConvBlock_34428457845562
// MI455X (gfx1250) — compile-verified
//
#include <hip/hip_runtime.h>
#include <hip/hip_bf16.h>
#include <math.h>

typedef __attribute__((ext_vector_type(2))) float v2f;
typedef __attribute__((ext_vector_type(8))) float v8f;

#define C_CH   64
#define S_SLOT 8
#define YSTRIDE 576   // 8*64 spline cols + 64 root cols
#define EPSV 1e-5f

// ---------------------------------------------------------------------------
// Zero accumulators (agg, cnt, sums) — must run every launch (atomics follow).
// ---------------------------------------------------------------------------
__global__ void k_zero(float* __restrict__ p, long long n) {
  long long i = (long long)blockIdx.x * blockDim.x + threadIdx.x;
  if (i < n) p[i] = 0.0f;
}

// ---------------------------------------------------------------------------
// Node-level GEMM with fp32 WMMA:
//   Y[n, t*16 + j] for t in [0,32): (x @ W[s])  with s = t>>2, cout = (t&3)*16+j
//   Y[n, 512 + c]               : (x @ root)
// One wave per 16-row tile; A tile (16x64) preloaded into registers
// (16 x float2 per lane, per the 16x4 f32 A-matrix VGPR layout), then all 36
// 16-wide column tiles are computed with 16 chained v_wmma_f32_16x16x4_f32.
// Store path is wave-uniform (full tile) to avoid per-store EXEC churn.
// ---------------------------------------------------------------------------
__global__ void k_gemm_wmma(const float* __restrict__ X,
                            const float* __restrict__ W,     // [8,64,64]
                            const float* __restrict__ root,  // [64,64]
                            float* __restrict__ Y,           // [N,576]
                            int N, int tiles_m) {
  const int wave = (int)(((long long)blockIdx.x * blockDim.x + threadIdx.x) >> 5);
  if (wave >= tiles_m) return;                 // wave-uniform exit (EXEC all-1s for WMMA)
  const int lane = threadIdx.x & 31;
  const int half = lane >> 4;                  // 0: K=0,1   1: K=2,3
  const int lm   = lane & 15;
  const int m0   = wave * 16;
  const int row  = min(m0 + lm, N - 1);        // clamp loads only (tail tile)
  const bool full_tile = (m0 + 16 <= N);       // wave-uniform

  // Preload A tile: a[kk] covers K = 4*kk + 2*half + {0,1}
  v2f a[16];
  const float* xrow = X + (long long)row * C_CH + 2 * half;
#pragma unroll
  for (int kk = 0; kk < 16; ++kk)
    a[kk] = *(const v2f*)(xrow + kk * 4);

  for (int t = 0; t < 36; ++t) {
    // B element B[k][j] lives at bcol[k*64]; this lane needs rows 4*kk+2*half(+1), col lm
    const float* bcol = (t < 32)
        ? (W + (t >> 2) * (C_CH * C_CH) + (t & 3) * 16 + lm + 2 * half * C_CH)
        : (root + (t - 32) * 16 + lm + 2 * half * C_CH);

    v8f acc = {};
#pragma unroll
    for (int kk = 0; kk < 16; ++kk) {
      v2f b;
      b.x = bcol[kk * 256];        // row 4*kk + 2*half
      b.y = bcol[kk * 256 + 64];   // row 4*kk + 2*half + 1
      acc = __builtin_amdgcn_wmma_f32_16x16x4_f32(
          /*neg_a=*/false, a[kk], /*neg_b=*/false, b,
          /*c_mod=*/(short)0, acc, /*reuse_a=*/false, /*reuse_b=*/false);
    }
    // D layout: vgpr i -> M = 8*half + i, N-col = lm
    float* yout = Y + (long long)(m0 + half * 8) * YSTRIDE + t * 16 + lm;
    if (full_tile) {
#pragma unroll
      for (int i = 0; i < 8; ++i) yout[(long long)i * YSTRIDE] = acc[i];
    } else {
#pragma unroll
      for (int i = 0; i < 8; ++i) {
        int r = m0 + half * 8 + i;
        if (r < N) Y[(long long)r * YSTRIDE + t * 16 + lm] = acc[i];
      }
    }
  }
}

// ---------------------------------------------------------------------------
// Edge phase: per (edge, channel) thread. Recompute the 8 spline (weight,slot)
// pairs, gather 8 rows of Y[src] (all L2-resident), scatter-add to agg[dst].
// ---------------------------------------------------------------------------
__global__ void k_edge(const float* __restrict__ Y,
                       const float* __restrict__ edge_attr,  // [E,3]
                       const int*   __restrict__ edge_index, // [2,E]
                       float* __restrict__ agg,              // [N,64]
                       float* __restrict__ cnt,              // [N]
                       int E) {
  long long gid = (long long)blockIdx.x * blockDim.x + threadIdx.x;
  int e = (int)(gid >> 6);
  if (e >= E) return;
  int c = (int)(gid & 63);

  int src = edge_index[e];
  int dst = edge_index[E + e];

  // v = attr*(K-1) = attr  (K=2)
  float f0 = edge_attr[e * 3 + 0];
  float f1 = edge_attr[e * 3 + 1];
  float f2 = edge_attr[e * 3 + 2];
  float l0 = floorf(f0), l1 = floorf(f1), l2 = floorf(f2);
  float r0 = f0 - l0, r1 = f1 - l1, r2 = f2 - l2;
  int k0 = (int)l0, k1 = (int)l1, k2 = (int)l2;

  const float* ybase = Y + (long long)src * YSTRIDE + c;
  float acc = 0.0f;
#pragma unroll
  for (int s = 0; s < 8; ++s) {
    // bits per unravel_index(s,(2,2,2)): dim0=s>>2, dim1=(s>>1)&1, dim2=s&1
    int b0 = (s >> 2) & 1, b1 = (s >> 1) & 1, b2 = s & 1;
    float w = (b0 ? r0 : 1.0f - r0) * (b1 ? r1 : 1.0f - r1) * (b2 ? r2 : 1.0f - r2);
    int i0 = min(max(k0 + b0, 0), 1);
    int i1 = min(max(k1 + b1, 0), 1);
    int i2 = min(max(k2 + b2, 0), 1);
    int slot = i0 + 2 * i1 + 4 * i2;     // radix = K^arange(D) = [1,2,4]
    acc = fmaf(w, ybase[slot * C_CH], acc);
  }
  atomicAdd(agg + (long long)dst * C_CH + c, acc);
  if (c == 0) atomicAdd(cnt + dst, 1.0f);
}

// ---------------------------------------------------------------------------
// out = agg / max(cnt,1) + (x @ root)   (root part lives in Y cols 512..575)
// ---------------------------------------------------------------------------
__global__ void k_combine(const float* __restrict__ agg,
                          const float* __restrict__ cnt,
                          const float* __restrict__ Y,
                          float* __restrict__ out, int N) {
  long long gid = (long long)blockIdx.x * blockDim.x + threadIdx.x;
  if (gid >= (long long)N * C_CH) return;
  int n = (int)(gid >> 6), c = (int)(gid & 63);
  float d = fmaxf(cnt[n], 1.0f);
  out[gid] = agg[gid] / d + Y[(long long)n * YSTRIDE + 512 + c];
}

// ---------------------------------------------------------------------------
// Per-channel sum / sum-of-squares (coalesced rows, shared reduce, few atomics)
// ---------------------------------------------------------------------------
__global__ void k_stats(const float* __restrict__ out, float* __restrict__ sums,
                        int N) {
  __shared__ float shs[C_CH], shq[C_CH];
  int tid = threadIdx.x;
  int c = tid & 63, rl = tid >> 6;                 // 4 rows x 64 ch per iter
  if (tid < C_CH) { shs[tid] = 0.0f; shq[tid] = 0.0f; }
  __syncthreads();
  float s = 0.0f, q = 0.0f;
  for (int row = blockIdx.x * 4 + rl; row < N; row += gridDim.x * 4) {
    float v = out[(long long)row * C_CH + c];
    s += v; q = fmaf(v, v, q);
  }
  atomicAdd(&shs[c], s);
  atomicAdd(&shq[c], q);
  __syncthreads();
  if (tid < C_CH) {
    atomicAdd(&sums[tid], shs[tid]);
    atomicAdd(&sums[C_CH + tid], shq[tid]);
  }
}

// ---------------------------------------------------------------------------
// BatchNorm (population stats) + ELU, in place on d_out
// ---------------------------------------------------------------------------
__global__ void k_norm_elu(float* __restrict__ out, const float* __restrict__ sums,
                           const float* __restrict__ gamma,
                           const float* __restrict__ beta, int N) {
  long long gid = (long long)blockIdx.x * blockDim.x + threadIdx.x;
  if (gid >= (long long)N * C_CH) return;
  int c = (int)(gid & 63);
  float invN = 1.0f / (float)N;
  float mean = sums[c] * invN;
  float var  = sums[C_CH + c] * invN - mean * mean;
  float rstd = rsqrtf(var + EPSV);
  float y = (out[gid] - mean) * rstd * gamma[c] + beta[c];
  out[gid] = y > 0.0f ? y : expm1f(y);
}

// ---------------------------------------------------------------------------
extern "C" void kernel_launch(void* const* d_in, const int* in_sizes, int n_in,
                              void* d_out, int out_size, void* d_ws, size_t ws_size,
                              hipStream_t stream) {
  const float* x         = (const float*)d_in[0];   // [N,64]
  const float* edge_attr = (const float*)d_in[1];   // [E,3]
  const float* W         = (const float*)d_in[2];   // [8,64,64]
  const float* root      = (const float*)d_in[3];   // [64,64]
  const float* gamma     = (const float*)d_in[4];   // [64]
  const float* beta      = (const float*)d_in[5];   // [64]
  const int*   edge_idx  = (const int*)d_in[6];     // [2,E]

  const int N = in_sizes[0] / C_CH;
  const int E = in_sizes[6] / 2;
  float* out = (float*)d_out;

  // workspace layout
  float* Y    = (float*)d_ws;                       // N*576
  float* agg  = Y + (long long)N * YSTRIDE;         // N*64
  float* cnt  = agg + (long long)N * C_CH;          // N
  float* sums = cnt + N;                            // 128

  // 1) zero accumulators
  long long nz = (long long)N * C_CH + N + 2 * C_CH;
  k_zero<<<(int)((nz + 255) / 256), 256, 0, stream>>>(agg, nz);

  // 2) node GEMM (fp32 WMMA): Y = X @ [W_0..W_7 | root]
  int tiles_m = (N + 15) / 16;
  int gemm_blocks = (tiles_m * 32 + 255) / 256;     // 8 waves / block
  k_gemm_wmma<<<gemm_blocks, 256, 0, stream>>>(x, W, root, Y, N, tiles_m);

  // 3) edge gather + scatter-add
  long long ethreads = (long long)E * C_CH;
  k_edge<<<(int)((ethreads + 255) / 256), 256, 0, stream>>>(
      Y, edge_attr, edge_idx, agg, cnt, E);

  // 4) combine: mean aggregation + root term
  long long nthreads = (long long)N * C_CH;
  k_combine<<<(int)((nthreads + 255) / 256), 256, 0, stream>>>(agg, cnt, Y, out, N);

  // 5) channel statistics
  k_stats<<<128, 256, 0, stream>>>(out, sums, N);

  // 6) batchnorm + ELU in place
  k_norm_elu<<<(int)((nthreads + 255) / 256), 256, 0, stream>>>(out, sums, gamma, beta, N);
}